// SparseArchShark_13838384628036
// MI455X (gfx1250) — compile-verified
//
#include <hip/hip_runtime.h>

// EmbeddingBag-sum for T tables: out[b][t][d] = sum_{i in bag(t,b)} weights[t][values[t][i]][d]
// Bandwidth-bound random gather: ~0.9 GB moved -> ~40us floor at 23.3 TB/s.
// Strategy: wave32-native (1 wave = 1 bag, 32 lanes x float4 = one 128-float row),
// global_prefetch_b8 to put a whole 32-row chunk (16KB, up to 128 cachelines) in
// flight with zero LOADcnt cost, readlane for scalar row indices, and an 8-deep
// unrolled global_load_b128 gather/accumulate for per-lane MLP.

static constexpr int kT     = 13;      // tables
static constexpr int kL     = 131072;  // lookups per table
static constexpr int kB     = 4096;    // bags per table
static constexpr int kN     = 100000;  // rows per table
static constexpr int kD     = 128;     // embedding dim
static constexpr int kLanes = 32;      // wave32
static constexpr int kWaves = 8;       // waves per block (256 threads)

__global__ __launch_bounds__(kWaves * kLanes)
void SparseArchShark_embbag_kernel(const int* __restrict__ values,
                                   const int* __restrict__ offsets,
                                   const float* __restrict__ weights,
                                   float* __restrict__ out) {
  const int lane = (int)(threadIdx.x & (kLanes - 1));
  const int wave = (int)(threadIdx.x >> 5);
  const int task = (int)blockIdx.x * kWaves + wave;   // one wave per (table,bag)
  if (task >= kT * kB) return;                        // whole wave exits together

  const int t = task >> 12;           // task / 4096
  const int b = task & (kB - 1);      // task % 4096

  // Bag range [begin, end): offsets are sorted, offsets[0]==0, last bag ends at L.
  const int* offs  = offsets + t * kB;
  const int  begin = offs[b];
  const int  end   = (b == kB - 1) ? kL : offs[b + 1];

  const int*   vals = values  + (size_t)t * kL;
  const float* wt   = weights + (size_t)t * ((size_t)kN * kD);

  float ax = 0.f, ay = 0.f, az = 0.f, aw = 0.f;

  for (int cs = begin; cs < end; cs += kLanes) {
    const int cnt = min(kLanes, end - cs);

    // Coalesced chunk of up to 32 row indices, one per lane.
    int myIdx = 0;
    if (lane < cnt) {
      myIdx = vals[cs + lane];
      // Each lane prefetches the 4x128B cachelines of ITS row: 4 instructions
      // per wave put the entire chunk in flight before the consume loop.
      const char* p = (const char*)(wt + (size_t)myIdx * kD);
      __builtin_prefetch(p,       0, 1);
      __builtin_prefetch(p + 128, 0, 1);
      __builtin_prefetch(p + 256, 0, 1);
      __builtin_prefetch(p + 384, 0, 1);
    }

    int j = 0;
    // 8-deep unroll: 8 global_load_b128 outstanding per lane before the adds.
    for (; j + 8 <= cnt; j += 8) {
      // Loop counter is wave-uniform -> readlane yields scalar (SGPR) row index.
      const int r0 = __builtin_amdgcn_readlane(myIdx, j + 0);
      const int r1 = __builtin_amdgcn_readlane(myIdx, j + 1);
      const int r2 = __builtin_amdgcn_readlane(myIdx, j + 2);
      const int r3 = __builtin_amdgcn_readlane(myIdx, j + 3);
      const int r4 = __builtin_amdgcn_readlane(myIdx, j + 4);
      const int r5 = __builtin_amdgcn_readlane(myIdx, j + 5);
      const int r6 = __builtin_amdgcn_readlane(myIdx, j + 6);
      const int r7 = __builtin_amdgcn_readlane(myIdx, j + 7);
      const float4 v0 = *((const float4*)(wt + (size_t)r0 * kD) + lane);
      const float4 v1 = *((const float4*)(wt + (size_t)r1 * kD) + lane);
      const float4 v2 = *((const float4*)(wt + (size_t)r2 * kD) + lane);
      const float4 v3 = *((const float4*)(wt + (size_t)r3 * kD) + lane);
      const float4 v4 = *((const float4*)(wt + (size_t)r4 * kD) + lane);
      const float4 v5 = *((const float4*)(wt + (size_t)r5 * kD) + lane);
      const float4 v6 = *((const float4*)(wt + (size_t)r6 * kD) + lane);
      const float4 v7 = *((const float4*)(wt + (size_t)r7 * kD) + lane);
      ax += ((v0.x + v1.x) + (v2.x + v3.x)) + ((v4.x + v5.x) + (v6.x + v7.x));
      ay += ((v0.y + v1.y) + (v2.y + v3.y)) + ((v4.y + v5.y) + (v6.y + v7.y));
      az += ((v0.z + v1.z) + (v2.z + v3.z)) + ((v4.z + v5.z) + (v6.z + v7.z));
      aw += ((v0.w + v1.w) + (v2.w + v3.w)) + ((v4.w + v5.w) + (v6.w + v7.w));
    }
    // 4-deep tail.
    for (; j + 4 <= cnt; j += 4) {
      const int r0 = __builtin_amdgcn_readlane(myIdx, j + 0);
      const int r1 = __builtin_amdgcn_readlane(myIdx, j + 1);
      const int r2 = __builtin_amdgcn_readlane(myIdx, j + 2);
      const int r3 = __builtin_amdgcn_readlane(myIdx, j + 3);
      const float4 v0 = *((const float4*)(wt + (size_t)r0 * kD) + lane);
      const float4 v1 = *((const float4*)(wt + (size_t)r1 * kD) + lane);
      const float4 v2 = *((const float4*)(wt + (size_t)r2 * kD) + lane);
      const float4 v3 = *((const float4*)(wt + (size_t)r3 * kD) + lane);
      ax += (v0.x + v1.x) + (v2.x + v3.x);
      ay += (v0.y + v1.y) + (v2.y + v3.y);
      az += (v0.z + v1.z) + (v2.z + v3.z);
      aw += (v0.w + v1.w) + (v2.w + v3.w);
    }
    // Scalar tail.
    for (; j < cnt; ++j) {
      const int r = __builtin_amdgcn_readlane(myIdx, j);
      const float4 v = *((const float4*)(wt + (size_t)r * kD) + lane);
      ax += v.x; ay += v.y; az += v.z; aw += v.w;
    }
  }

  // Output [B, T, D]; wave writes one contiguous 512B row. Empty bags write zeros.
  float4 res; res.x = ax; res.y = ay; res.z = az; res.w = aw;
  *((float4*)(out + ((size_t)b * kT + t) * kD) + lane) = res;
}

extern "C" void kernel_launch(void* const* d_in, const int* in_sizes, int n_in,
                              void* d_out, int out_size, void* d_ws, size_t ws_size,
                              hipStream_t stream) {
  (void)in_sizes; (void)n_in; (void)d_ws; (void)ws_size; (void)out_size;
  const int*   values  = (const int*)d_in[0];    // [T, L] int32
  const int*   offsets = (const int*)d_in[1];    // [T, B] int32
  const float* weights = (const float*)d_in[2];  // [T, N, D] fp32
  float*       out     = (float*)d_out;          // [B, T, D] fp32

  const int tasks  = kT * kB;                       // 53248 waves
  const int blocks = (tasks + kWaves - 1) / kWaves; // 6656 blocks of 256
  SparseArchShark_embbag_kernel<<<blocks, kWaves * kLanes, 0, stream>>>(
      values, offsets, weights, out);
}